// CRF_17420387353216
// MI455X (gfx1250) — compile-verified
//
#include <hip/hip_runtime.h>

#define B_ 256
#define S_ 512
#define T_ 32
#define NSTEP 510   /* S_ - 2 sequential recursion steps */

typedef __attribute__((ext_vector_type(2))) float v2f;
typedef __attribute__((ext_vector_type(8))) float v8f;

__device__ __forceinline__ float bshflx(float v, int m)   { return __shfl_xor(v, m, 32); }
__device__ __forceinline__ int   ishflx(int v, int m)     { return __shfl_xor(v, m, 32); }

// Broadcast lane `src` (compile-time-ish uniform) -> v_readlane_b32, no DS op.
__device__ __forceinline__ float rdlane(float v, int src) {
  return __int_as_float(__builtin_amdgcn_readlane(__float_as_int(v), src));
}

// xor-16 half-wave swap via v_permlanex16_b32 (identity selects), no DS op.
__device__ __forceinline__ float xor16(float v) {
  int i = __float_as_int(v);
  int r = __builtin_amdgcn_permlanex16(i, i, 0x76543210, 0xfedcba98, false, false);
  return __int_as_float(r);
}

// ---------------------------------------------------------------------------
// Kernel 1: NLL forward partition (logsumexp recursion) as exp-space GEMM.
// One wave32 per 16 batches. State kept in WMMA A-fragment layout:
//   lane l (lo=l&15, hi=l>>4) holds p[b=bg+lo][i], i = 4*kb + r + 2*hi
// Step:  p_new = f_t + m + log( exp(p - m) @ exp(transitions) )
// The 16x32 @ 32x32 product uses v_wmma_f32_16x16x4_f32 (8 K-tiles x 2 N-tiles).
// ---------------------------------------------------------------------------
__global__ __launch_bounds__(32) void crf_fwd_lse(
    const float* __restrict__ feats, const float* __restrict__ trans,
    const unsigned char* __restrict__ mask, float* __restrict__ fwd_part)
{
  const int l  = threadIdx.x;
  const int lo = l & 15;
  const int hi = l >> 4;
  const int bg = blockIdx.x << 4;           // first batch of this group

  // ---- sequence length of this lane's batch (bg+lo); mask is a prefix mask
  int len;
  {
    const unsigned int* mrow =
        (const unsigned int*)(mask + (size_t)(bg + lo) * S_) + hi * 64;
    int s = 0;
    for (int q = 0; q < 64; ++q) s += __popc(mrow[q] & 0x01010101u);
    len = s + ishflx(s, 16);
  }

  // ---- exp(transitions) as constant B-fragments (layout mirrors A fragments)
  float etB[2][8][2];
  #pragma unroll
  for (int tile = 0; tile < 2; ++tile)
    #pragma unroll
    for (int kb = 0; kb < 8; ++kb)
      #pragma unroll
      for (int r = 0; r < 2; ++r) {
        int i = 4 * kb + r + 2 * hi;        // row of transitions (K)
        int j = 16 * tile + lo;             // col of transitions (N)
        etB[tile][kb][r] = __expf(trans[i * T_ + j]);
      }

  // ---- partition state in A-fragment layout, init = feats[:,1,:]
  float pA[16];
  #pragma unroll
  for (int kb = 0; kb < 8; ++kb)
    #pragma unroll
    for (int r = 0; r < 2; ++r) {
      int i = 4 * kb + r + 2 * hi;
      pA[2 * kb + r] = feats[(size_t)(bg + lo) * S_ * T_ + 1 * T_ + i];
    }

  __shared__ float pbuf[16 * 32];           // D-layout -> A-layout transpose buffer

  // f_t in D-fragment layout (row b = d + 8*hi, col j = lo + 16*tile), prefetched
  float fD[16];
  #pragma unroll
  for (int d = 0; d < 8; ++d)
    #pragma unroll
    for (int tile = 0; tile < 2; ++tile)
      fD[d * 2 + tile] =
          feats[(size_t)(bg + d + 8 * hi) * S_ * T_ + 2 * T_ + lo + 16 * tile];

  for (int k = 0; k < NSTEP; ++k) {
    // prefetch next step's emissions (hides latency behind exp/WMMA/log chain)
    float fN[16];
    if (k < NSTEP - 1) {
      #pragma unroll
      for (int d = 0; d < 8; ++d)
        #pragma unroll
        for (int tile = 0; tile < 2; ++tile)
          fN[d * 2 + tile] = feats[(size_t)(bg + d + 8 * hi) * S_ * T_ +
                                   (size_t)(k + 3) * T_ + lo + 16 * tile];
    } else {
      #pragma unroll
      for (int q = 0; q < 16; ++q) fN[q] = 0.f;
    }

    // per-batch row max (balanced tree + permlanex16 half-swap)
    float mt[8];
    #pragma unroll
    for (int a = 0; a < 8; ++a) mt[a] = fmaxf(pA[a], pA[a + 8]);
    #pragma unroll
    for (int a = 0; a < 4; ++a) mt[a] = fmaxf(mt[a], mt[a + 4]);
    float m = fmaxf(fmaxf(mt[0], mt[1]), fmaxf(mt[2], mt[3]));
    m = fmaxf(m, xor16(m));

    // E = exp(p - m): already in A-fragment layout
    float ea[16];
    #pragma unroll
    for (int a = 0; a < 16; ++a) ea[a] = __expf(pA[a] - m);

    // [16x32] = [16x32] @ [32x32] on the matrix pipe
    v8f acc0 = {0.f, 0.f, 0.f, 0.f, 0.f, 0.f, 0.f, 0.f};
    v8f acc1 = {0.f, 0.f, 0.f, 0.f, 0.f, 0.f, 0.f, 0.f};
    #pragma unroll
    for (int kb = 0; kb < 8; ++kb) {
      v2f a2 = {ea[2 * kb], ea[2 * kb + 1]};
      v2f b0 = {etB[0][kb][0], etB[0][kb][1]};
      v2f b1 = {etB[1][kb][0], etB[1][kb][1]};
      acc0 = __builtin_amdgcn_wmma_f32_16x16x4_f32(false, a2, false, b0,
                                                   (short)0, acc0, false, false);
      acc1 = __builtin_amdgcn_wmma_f32_16x16x4_f32(false, a2, false, b1,
                                                   (short)0, acc1, false, false);
    }

    // p_new = f + m + log(acc), written in D layout to LDS.
    // m broadcast to D-layout rows via v_readlane (uniform lane indices).
    #pragma unroll
    for (int d = 0; d < 8; ++d) {
      float mlo = rdlane(m, d);
      float mhi = rdlane(m, d + 8);
      float mm  = hi ? mhi : mlo;           // batch row d + 8*hi
      float p0 = fD[d * 2 + 0] + mm + __logf(acc0[d]);
      float p1 = fD[d * 2 + 1] + mm + __logf(acc1[d]);
      pbuf[(d + 8 * hi) * 32 + lo]      = p0;
      pbuf[(d + 8 * hi) * 32 + lo + 16] = p1;
    }
    __syncthreads();

    // read back in A layout; freeze batches whose sequence ended (mask prefix)
    bool act = (k < len - 2);
    #pragma unroll
    for (int kb = 0; kb < 8; ++kb)
      #pragma unroll
      for (int r = 0; r < 2; ++r) {
        int i = 4 * kb + r + 2 * hi;
        float nv = pbuf[lo * 32 + i];
        pA[2 * kb + r] = act ? nv : pA[2 * kb + r];
      }
    __syncthreads();

    #pragma unroll
    for (int q = 0; q < 16; ++q) fD[q] = fN[q];
  }

  // final logsumexp over tags, then sum over this group's 16 batches
  float mx = pA[0];
  #pragma unroll
  for (int a = 1; a < 16; ++a) mx = fmaxf(mx, pA[a]);
  mx = fmaxf(mx, xor16(mx));
  float se = 0.f;
  #pragma unroll
  for (int a = 0; a < 16; ++a) se += __expf(pA[a] - mx);
  se += xor16(se);
  float lse = mx + __logf(se);              // replicated at lanes lo and lo+16
  float contrib = (hi == 0) ? lse : 0.f;
  #pragma unroll
  for (int off = 16; off > 0; off >>= 1) contrib += bshflx(contrib, off);
  if (l == 0) fwd_part[blockIdx.x] = contrib;   // deterministic partial
}

// ---------------------------------------------------------------------------
// Kernel 2: Viterbi (max-plus recursion, LDS-resident backpointers, backtrack)
// + gold path score partial. One wave32 per batch, lane = tag.
// Broadcasts use v_readlane (constant lanes); argmax is a register tournament
// tree with exact first-occurrence tie-break (matches jnp.argmax).
// ---------------------------------------------------------------------------
__global__ __launch_bounds__(32) void crf_viterbi(
    const float* __restrict__ feats, const float* __restrict__ trans,
    const unsigned char* __restrict__ mask, const int* __restrict__ tags,
    float* __restrict__ gold_part, float* __restrict__ out)
{
  __shared__ unsigned char bpbuf[NSTEP * 32];   // 16320 B backpointers
  const int b = blockIdx.x;
  const int j = threadIdx.x;

  // sequence length (count of nonzero mask bytes in this row)
  const unsigned int* mrow = (const unsigned int*)(mask + (size_t)b * S_);
  int s = 0;
  #pragma unroll
  for (int q = 0; q < 4; ++q) s += __popc(mrow[j * 4 + q] & 0x01010101u);
  #pragma unroll
  for (int off = 16; off > 0; off >>= 1) s += ishflx(s, off);
  const int len = s;                       // uniform across the wave

  float trcol[32];                         // column j of transitions
  #pragma unroll
  for (int i = 0; i < 32; ++i) trcol[i] = trans[i * T_ + j];

  const float* fbase = feats + (size_t)b * S_ * T_;
  float v = fbase[1 * T_ + j];             // part0 = feats[:,1,:]
  float lastp = v;

  for (int k = 0; k < NSTEP; ++k) {
    if (k == len - 2) lastp = v;           // history[len-2] (state BEFORE step k)
    // stream the emission rows ahead of the chain
    {
      int pt = (k + 10 < S_) ? (k + 10) : (S_ - 1);
      __builtin_prefetch(fbase + (size_t)pt * T_ + j, 0, 1);
    }
    float f = fbase[(size_t)(k + 2) * T_ + j];

    // candidates: c[i] = v[i] + trans[i][j]  (v_readlane broadcast, no DS)
    float cv[32];
    int   ci[32];
    #pragma unroll
    for (int i = 0; i < 32; ++i) {
      cv[i] = rdlane(v, i) + trcol[i];
      ci[i] = i;
    }
    // tournament tree, first-occurrence tie-break (lower index wins ties)
    #pragma unroll
    for (int w = 16; w >= 1; w >>= 1)
      #pragma unroll
      for (int i = 0; i < 16; ++i) {
        if (i < w) {
          bool take = (cv[i + w] > cv[i]) ||
                      ((cv[i + w] == cv[i]) && (ci[i + w] < ci[i]));
          cv[i] = take ? cv[i + w] : cv[i];
          ci[i] = take ? ci[i + w] : ci[i];
        }
      }
    float best = cv[0];
    int   bi   = ci[0];

    bool mk = (k + 2) < len;               // mask[b, k+2]
    bpbuf[k * 32 + j] = (unsigned char)(mk ? bi : 0);
    v = f + best;                          // partition always advances
  }
  if (len - 2 == NSTEP) lastp = v;         // full-length sequences

  // pointer = first-index argmax_j lastp
  float av = lastp;
  int ai = j;
  #pragma unroll
  for (int off = 16; off > 0; off >>= 1) {
    float ov = bshflx(av, off);
    int oi = ishflx(ai, off);
    bool take = (ov > av) || (ov == av && oi < ai);
    av = take ? ov : av;
    ai = take ? oi : ai;
  }
  const int pointer = ai;                  // uniform

  // gold score partial for this batch
  float g = 0.f;
  for (int ss = 1 + j; ss < S_; ss += 32) {
    if (ss < len) {
      int tg = tags[(size_t)b * S_ + ss];
      tg = (tg == -100) ? 0 : tg;
      float e = fbase[(size_t)ss * T_ + tg];
      if (ss >= 2) {
        int tp = tags[(size_t)b * S_ + ss - 1];
        tp = (tp == -100) ? 0 : tp;
        e += trans[tp * T_ + tg];
      }
      g += e;
    }
  }
  #pragma unroll
  for (int off = 16; off > 0; off >>= 1) g += bshflx(g, off);
  if (j == 0) gold_part[b] = g;

  // outputs: path_score (zeros) + decode row; LDS chain walk (wave-local)
  float* dec = out + 1 + B_ + (size_t)b * S_;
  if (j == 0) {
    out[1 + b] = 0.f;                      // path_score
    dec[0] = 0.f;
    dec[S_ - 1] = (float)pointer;
    int ptr = pointer;
    for (int k = NSTEP - 1; k >= 0; --k) { // exact reference backtrack
      ptr = (int)bpbuf[k * 32 + ptr];
      dec[k + 1] = (float)ptr;
    }
  }
}

// ---------------------------------------------------------------------------
// Kernel 3: deterministic final reduction -> loss
// ---------------------------------------------------------------------------
__global__ void crf_finalize(const float* __restrict__ fwd_part,
                             const float* __restrict__ gold_part,
                             float* __restrict__ out)
{
  if (threadIdx.x == 0 && blockIdx.x == 0) {
    float f = 0.f;
    for (int i = 0; i < 16; ++i) f += fwd_part[i];
    float g = 0.f;
    for (int i = 0; i < B_; ++i) g += gold_part[i];
    out[0] = (f - g) / (float)B_;
  }
}

extern "C" void kernel_launch(void* const* d_in, const int* in_sizes, int n_in,
                              void* d_out, int out_size, void* d_ws, size_t ws_size,
                              hipStream_t stream)
{
  const float*         feats = (const float*)d_in[0];
  const float*         trans = (const float*)d_in[1];
  const unsigned char* mask  = (const unsigned char*)d_in[2];  // jnp bool = 1 byte
  const int*           tags  = (const int*)d_in[3];
  float* out = (float*)d_out;          // [0]=loss, [1..256]=path_score, rest=decode

  float* fwd_part  = (float*)d_ws;     // 16 partials
  float* gold_part = fwd_part + 16;    // 256 partials

  crf_fwd_lse <<<B_ / 16, 32, 0, stream>>>(feats, trans, mask, fwd_part);
  crf_viterbi <<<B_,      32, 0, stream>>>(feats, trans, mask, tags, gold_part, out);
  crf_finalize<<<1,       32, 0, stream>>>(fwd_part, gold_part, out);
}